// NodeEdgeBlockImproved_19713899889387
// MI455X (gfx1250) — compile-verified
//
#include <hip/hip_runtime.h>
#include <hip/hip_bf16.h>

// ---------------- problem constants ----------------
#define BS 4
#define NN 192           // nodes
#define DX 256
#define DE 64
#define DY 64
#define NHEAD 8
#define DF 32            // DX / NHEAD
#define ROWS (BS*NN)             // 768
#define EROWS (BS*NN*NN)         // 147456 flattened edges
#define INV_SQRT_DF 0.17677669529663687f

typedef __attribute__((ext_vector_type(2))) float v2f;
typedef __attribute__((ext_vector_type(8))) float v8f;
typedef __attribute__((ext_vector_type(4))) unsigned int u32x4;
typedef __attribute__((ext_vector_type(8))) int i32x8;
typedef __attribute__((ext_vector_type(4))) int i32x4;

// Tensor Data Mover availability (this toolchain: clang-23, 6-arg builtin)
#if defined(__has_builtin)
#if __has_builtin(__builtin_amdgcn_tensor_load_to_lds) && \
    __has_builtin(__builtin_amdgcn_s_wait_tensorcnt)
#define USE_TDM 1
#endif
#endif
#ifndef USE_TDM
#define USE_TDM 0
#endif

// ---------------- WMMA helpers (wave32, 16x16x4 f32) ----------------
static __device__ __forceinline__ v8f wmma4(v2f a, v2f b, v8f c) {
  return __builtin_amdgcn_wmma_f32_16x16x4_f32(false, a, false, b, (short)0, c,
                                               false, false);
}

// A fragment: p points at element (m0, k0); element (m,k) at p[m*sm + k*sk]
static __device__ __forceinline__ v2f ldA(const float* p, int sm, int sk) {
  int lane = threadIdx.x & 31;
  int m = lane & 15;
  int k = (lane >> 4) << 1;
  v2f a;
  a.x = p[m * sm + k * sk];
  a.y = p[m * sm + (k + 1) * sk];
  return a;
}

// B fragment: p points at element (k0, n0); element (k,n) at p[k*sk + n*sn]
static __device__ __forceinline__ v2f ldB(const float* p, int sk, int sn) {
  int lane = threadIdx.x & 31;
  int n = lane & 15;
  int k = (lane >> 4) << 1;
  v2f b;
  b.x = p[k * sk + n * sn];
  b.y = p[(k + 1) * sk + n * sn];
  return b;
}

// ---------------- kernel 1: Q/K/V = (X @ W + b) * x_mask ----------------
__global__ void __launch_bounds__(32)
qkv_kernel(const float* __restrict__ X,
           const float* __restrict__ qW, const float* __restrict__ qb,
           const float* __restrict__ kW, const float* __restrict__ kb,
           const float* __restrict__ vW, const float* __restrict__ vb,
           const unsigned char* __restrict__ mask,
           float* __restrict__ Q, float* __restrict__ Km, float* __restrict__ Vm) {
  const float* W; const float* bias; float* Out;
  if (blockIdx.z == 0)      { W = qW; bias = qb; Out = Q; }
  else if (blockIdx.z == 1) { W = kW; bias = kb; Out = Km; }
  else                      { W = vW; bias = vb; Out = Vm; }
  int m0 = blockIdx.x * 16, n0 = blockIdx.y * 16;
  int lane = threadIdx.x;
  v8f c = {};
  for (int k0 = 0; k0 < DX; k0 += 4) {
    v2f a = ldA(X + (size_t)m0 * DX + k0, DX, 1);
    v2f b = ldB(W + (size_t)k0 * DX + n0, DX, 1);
    c = wmma4(a, b, c);
  }
  int col = n0 + (lane & 15);
  float bv = bias[col];
#pragma unroll
  for (int v = 0; v < 8; v++) {
    int row = m0 + ((lane >> 4) << 3) + v;
    float xm = mask[row] ? 1.f : 0.f;
    Out[(size_t)row * DX + col] = (c[v] + bv) * xm;
  }
}

// ---------------- kernel 2: fused edge pipeline ----------------
// Per block: 16 flattened edges. E-tile -> LDS via TDM; E1/E2 via WMMA with
// register-cached A fragments; Y stays in LDS; per-head score sums;
// newE = Y @ eoutW via WMMA. Y never touches HBM.
__global__ void __launch_bounds__(32)
edge_kernel(const float* __restrict__ E,
            const float* __restrict__ Q, const float* __restrict__ Km,
            const float* __restrict__ emulW, const float* __restrict__ emulb,
            const float* __restrict__ eaddW, const float* __restrict__ eaddb,
            const float* __restrict__ eoutW, const float* __restrict__ eoutb,
            const unsigned char* __restrict__ mask,
            float* __restrict__ scores, float* __restrict__ newE) {
  __shared__ float Et[16 * DE];   // 4 KB  (one TDM tile)
  __shared__ float Yb[16 * DX];   // 16 KB
  int r0 = blockIdx.x * 16;
  int lane = threadIdx.x;

  // warm L2 for the shared edge-path weights
  __builtin_prefetch(emulW, 0, 1);
  __builtin_prefetch(eaddW, 0, 1);
  __builtin_prefetch(eoutW, 0, 1);

#if USE_TDM
  {
    // 1-D tensor DMA: 1024 f32 elements (16 rows x 64) global -> LDS.
    unsigned long long ga = (unsigned long long)(const void*)(E + (size_t)r0 * DE);
    unsigned ldsa = (unsigned)(unsigned long long)(const void*)&Et[0];
    u32x4 g0;
    g0.x = 1u;                                            // count=1 valid D#
    g0.y = ldsa;                                          // lds_addr
    g0.z = (unsigned)(ga & 0xffffffffull);                // global_addr[31:0]
    g0.w = (unsigned)((ga >> 32) & 0x1ffffffull)          // global_addr[56:32]
           | (2u << 30);                                  // type = 2 (image)
    i32x8 g1;
    g1[0] = (int)(2u << 16);        // workgroup_mask=0, data_size=4B
    g1[1] = (int)(1024u << 16);     // tensor_dim0 = 1024 (bits 79:48, low half)
    g1[2] = (int)(1u << 16);        // tensor_dim0 hi=0, tensor_dim1 = 1
    g1[3] = (int)(1024u << 16);     // tile_dim0 = 1024 (bits 127:112)
    g1[4] = 0;                      // tile_dim1 = 0 (1-D tile), tile_dim2 = 0
    g1[5] = 1024;                   // tensor_dim0_stride = 1024
    g1[6] = 0;
    g1[7] = 0;
    i32x4 z4; z4[0] = 0; z4[1] = 0; z4[2] = 0; z4[3] = 0;
    i32x8 z8; z8[0] = 0; z8[1] = 0; z8[2] = 0; z8[3] = 0;
    z8[4] = 0; z8[5] = 0; z8[6] = 0; z8[7] = 0;
    __builtin_amdgcn_tensor_load_to_lds(g0, g1, z4, z4, z8, 0);
    __builtin_amdgcn_s_wait_tensorcnt(0);
  }
#else
  for (int idx = lane; idx < 16 * DE; idx += 32)
    Et[idx] = E[(size_t)r0 * DE + idx];
#endif
  __syncthreads();

  // register-cache all 16 K-step A fragments of the E tile (32 VGPRs)
  v2f afrag[DE / 4];
#pragma unroll
  for (int kk = 0; kk < DE / 4; kk++) afrag[kk] = ldA(Et + kk * 4, DE, 1);

  for (int n0 = 0; n0 < DX; n0 += 16) {
    v8f c1 = {}; v8f c2 = {};
#pragma unroll
    for (int kk = 0; kk < DE / 4; kk++) {
      int k0 = kk * 4;
      v2f b1 = ldB(emulW + (size_t)k0 * DX + n0, DX, 1);
      c1 = wmma4(afrag[kk], b1, c1);
      v2f b2 = ldB(eaddW + (size_t)k0 * DX + n0, DX, 1);
      c2 = wmma4(afrag[kk], b2, c2);
    }
    int col = n0 + (lane & 15);
    float bm = emulb[col], ba = eaddb[col];
#pragma unroll
    for (int v = 0; v < 8; v++) {
      int rloc = ((lane >> 4) << 3) + v;
      int g = r0 + rloc;
      int b = g / (NN * NN);
      int rem = g - b * NN * NN;
      int i = rem / NN, j = rem - i * NN;
      float em = (mask[b * NN + i] && mask[b * NN + j]) ? 1.f : 0.f;
      float e1 = (c1[v] + bm) * em;
      float e2 = (c2[v] + ba) * em;
      float qv = Q[(size_t)(b * NN + i) * DX + col];
      float kv = Km[(size_t)(b * NN + j) * DX + col];
      Yb[rloc * DX + col] = qv * kv * INV_SQRT_DF * (e1 + 1.f) + e2;
    }
  }
  __syncthreads();

  // per-head score sums (Y.sum over df)
  for (int idx = lane; idx < 16 * NHEAD; idx += 32) {
    int rloc = idx >> 3, h = idx & 7;
    float s = 0.f;
#pragma unroll
    for (int d = 0; d < DF; d++) s += Yb[rloc * DX + h * DF + d];
    scores[(size_t)(r0 + rloc) * NHEAD + h] = s;
  }

  // newE = (Y @ eoutW + eoutb) * e_mask
  for (int n0 = 0; n0 < DE; n0 += 16) {
    v8f c = {};
    for (int k0 = 0; k0 < DX; k0 += 4) {
      v2f a = ldA(Yb + k0, DX, 1);
      v2f b = ldB(eoutW + (size_t)k0 * DE + n0, DE, 1);
      c = wmma4(a, b, c);
    }
    int col = n0 + (lane & 15);
    float bv = eoutb[col];
#pragma unroll
    for (int v = 0; v < 8; v++) {
      int rloc = ((lane >> 4) << 3) + v;
      int g = r0 + rloc;
      int b = g / (NN * NN);
      int rem = g - b * NN * NN;
      int i = rem / NN, j = rem - i * NN;
      float em = (mask[b * NN + i] && mask[b * NN + j]) ? 1.f : 0.f;
      newE[(size_t)g * DE + col] = (c[v] + bv) * em;
    }
  }
}

// ---------------- kernel 3: softmax over j, in place ----------------
__global__ void __launch_bounds__(32)
softmax_kernel(float* __restrict__ scores) {
  int idx = blockIdx.x;              // (b*NN + i)*NHEAD + h
  int h = idx & (NHEAD - 1);
  int bi = idx >> 3;
  float* base = scores + (size_t)bi * NN * NHEAD + h;
  int lane = threadIdx.x;
  float vals[6];
  float m = -3.4e38f;
#pragma unroll
  for (int t = 0; t < 6; t++) {
    int j = t * 32 + lane;
    vals[t] = base[(size_t)j * NHEAD];
    m = fmaxf(m, vals[t]);
  }
#pragma unroll
  for (int off = 16; off; off >>= 1) m = fmaxf(m, __shfl_xor(m, off, 32));
  float s = 0.f;
#pragma unroll
  for (int t = 0; t < 6; t++) { vals[t] = __expf(vals[t] - m); s += vals[t]; }
#pragma unroll
  for (int off = 16; off; off >>= 1) s += __shfl_xor(s, off, 32);
  float r = 1.f / s;
#pragma unroll
  for (int t = 0; t < 6; t++) {
    int j = t * 32 + lane;
    base[(size_t)j * NHEAD] = vals[t] * r;
  }
}

// ---------------- kernel 4: weighted_V = einsum('bijh,bjhd->bihd') ----------
__global__ void __launch_bounds__(32)
wv_kernel(const float* __restrict__ attn, const float* __restrict__ Vm,
          float* __restrict__ wV) {
  int m0 = blockIdx.x * 16;          // i tile (192/16 = 12)
  int n0 = blockIdx.y * 16;          // d tile (32/16 = 2)
  int bh = blockIdx.z;               // b*8 + h
  int b = bh >> 3, h = bh & 7;
  int lane = threadIdx.x;
  const float* Abase = attn + (size_t)b * NN * NN * NHEAD + h;   // [i*NN*NH + j*NH]
  const float* Bbase = Vm + (size_t)b * NN * DX + h * DF + n0;   // [k*DX + n]
  v8f c = {};
  for (int k0 = 0; k0 < NN; k0 += 4) {
    v2f a = ldA(Abase + (size_t)m0 * NN * NHEAD + (size_t)k0 * NHEAD,
                NN * NHEAD, NHEAD);
    v2f bb = ldB(Bbase + (size_t)k0 * DX, DX, 1);
    c = wmma4(a, bb, c);
  }
  int col = n0 + (lane & 15);
#pragma unroll
  for (int v = 0; v < 8; v++) {
    int row = m0 + ((lane >> 4) << 3) + v;
    wV[(size_t)(b * NN + row) * DX + h * DF + col] = c[v];
  }
}

// ---------------- kernel 5: yx1 = y@yxaW+yxab, yx2 = y@yxmW+yxmb ------------
__global__ void yx_kernel(const float* __restrict__ y,
                          const float* __restrict__ yxaW, const float* __restrict__ yxab,
                          const float* __restrict__ yxmW, const float* __restrict__ yxmb,
                          float* __restrict__ yx1, float* __restrict__ yx2) {
  int idx = blockIdx.x * blockDim.x + threadIdx.x;   // 0..2047
  if (idx >= 2 * BS * DX) return;
  int which = idx >> 10;
  int o = idx & 1023;
  int b = o >> 8, c = o & 255;
  const float* W = which ? yxmW : yxaW;
  const float* bias = which ? yxmb : yxab;
  float s = bias[c];
  for (int k = 0; k < DY; k++) s += y[b * DY + k] * W[k * DX + c];
  if (which) yx2[b * DX + c] = s; else yx1[b * DX + c] = s;
}

// ---------------- kernel 6: newX = ((yx1+(yx2+1)*wV) @ xoutW + b)*mask ------
__global__ void __launch_bounds__(32)
newx_kernel(const float* __restrict__ yx1, const float* __restrict__ yx2,
            const float* __restrict__ wV,
            const float* __restrict__ xoutW, const float* __restrict__ xoutb,
            const unsigned char* __restrict__ mask, float* __restrict__ newX) {
  int m0 = blockIdx.x * 16, n0 = blockIdx.y * 16;
  int lane = threadIdx.x;
  int bb = m0 / NN;                  // 192 % 16 == 0: tile within one batch
  v8f c = {};
  for (int k0 = 0; k0 < DX; k0 += 4) {
    int m = m0 + (lane & 15);
    int k = k0 + ((lane >> 4) << 1);
    v2f a;
    a.x = yx1[bb * DX + k]     + (yx2[bb * DX + k]     + 1.f) * wV[(size_t)m * DX + k];
    a.y = yx1[bb * DX + k + 1] + (yx2[bb * DX + k + 1] + 1.f) * wV[(size_t)m * DX + k + 1];
    v2f b = ldB(xoutW + (size_t)k0 * DX + n0, DX, 1);
    c = wmma4(a, b, c);
  }
  int col = n0 + (lane & 15);
  float bv = xoutb[col];
#pragma unroll
  for (int v = 0; v < 8; v++) {
    int row = m0 + ((lane >> 4) << 3) + v;
    float xm = mask[row] ? 1.f : 0.f;
    newX[(size_t)row * DX + col] = (c[v] + bv) * xm;
  }
}

// ---------------- kernel 7: zx stats over nodes (mean/min/max/std ddof=1) ---
__global__ void __launch_bounds__(32)
zx_kernel(const float* __restrict__ X, float* __restrict__ zx) {
  int idx = blockIdx.x;              // b*256 + c
  int b = idx >> 8, c = idx & 255;
  int lane = threadIdx.x;
  float s = 0.f, s2 = 0.f, mn = 3.4e38f, mx = -3.4e38f;
  for (int j = lane; j < NN; j += 32) {
    float v = X[(size_t)(b * NN + j) * DX + c];
    s += v; s2 += v * v; mn = fminf(mn, v); mx = fmaxf(mx, v);
  }
#pragma unroll
  for (int off = 16; off; off >>= 1) {
    s  += __shfl_xor(s, off, 32);
    s2 += __shfl_xor(s2, off, 32);
    mn = fminf(mn, __shfl_xor(mn, off, 32));
    mx = fmaxf(mx, __shfl_xor(mx, off, 32));
  }
  if (lane == 0) {
    float mean = s / (float)NN;
    float var = fmaxf(0.f, (s2 - (float)NN * mean * mean) / (float)(NN - 1));
    zx[b * 4 * DX + c]          = mean;
    zx[b * 4 * DX + DX + c]     = mn;
    zx[b * 4 * DX + 2 * DX + c] = mx;
    zx[b * 4 * DX + 3 * DX + c] = sqrtf(var);
  }
}

// ---------------- kernel 8: ze stats over all edges -------------------------
__global__ void __launch_bounds__(256)
ze_kernel(const float* __restrict__ E, float* __restrict__ ze) {
  int idx = blockIdx.x;              // b*64 + c
  int b = idx >> 6, c = idx & 63;
  int tid = threadIdx.x;
  float s = 0.f, s2 = 0.f, mn = 3.4e38f, mx = -3.4e38f;
  const int tot = NN * NN;
  for (int j = tid; j < tot; j += 256) {
    float v = E[((size_t)b * tot + j) * DE + c];
    s += v; s2 += v * v; mn = fminf(mn, v); mx = fmaxf(mx, v);
  }
#pragma unroll
  for (int off = 16; off; off >>= 1) {
    s  += __shfl_xor(s, off, 32);
    s2 += __shfl_xor(s2, off, 32);
    mn = fminf(mn, __shfl_xor(mn, off, 32));
    mx = fmaxf(mx, __shfl_xor(mx, off, 32));
  }
  __shared__ float ws[8], ws2[8], wmn[8], wmx[8];
  int w = tid >> 5, lane = tid & 31;
  if (lane == 0) { ws[w] = s; ws2[w] = s2; wmn[w] = mn; wmx[w] = mx; }
  __syncthreads();
  if (tid == 0) {
    s = 0.f; s2 = 0.f; mn = 3.4e38f; mx = -3.4e38f;
#pragma unroll
    for (int k = 0; k < 8; k++) {
      s += ws[k]; s2 += ws2[k]; mn = fminf(mn, wmn[k]); mx = fmaxf(mx, wmx[k]);
    }
    float mean = s / (float)tot;
    float var = fmaxf(0.f, (s2 - (float)tot * mean * mean) / (float)(tot - 1));
    ze[b * 4 * DE + c]          = mean;
    ze[b * 4 * DE + DE + c]     = mn;
    ze[b * 4 * DE + 2 * DE + c] = mx;
    ze[b * 4 * DE + 3 * DE + c] = sqrtf(var);
  }
}

// ---------------- kernel 9: y head (x_y, e_y, MLP) --------------------------
__global__ void __launch_bounds__(256)
yfinal_kernel(const float* __restrict__ y,
              const float* __restrict__ zx, const float* __restrict__ ze,
              const float* __restrict__ xyW, const float* __restrict__ xyb,
              const float* __restrict__ eyW, const float* __restrict__ eyb,
              const float* __restrict__ y1W, const float* __restrict__ y1b,
              const float* __restrict__ y2W, const float* __restrict__ y2b,
              float* __restrict__ outy) {
  __shared__ float T[BS * DY];
  __shared__ float H[BS * DY];
  int tid = threadIdx.x;             // 0..255 -> (b, c)
  int b = tid / DY, c = tid % DY;
  float xy = xyb[c];
  for (int k = 0; k < 4 * DX; k++) xy += zx[b * 4 * DX + k] * xyW[k * DY + c];
  float ey = eyb[c];
  for (int k = 0; k < 4 * DE; k++) ey += ze[b * 4 * DE + k] * eyW[k * DY + c];
  T[tid] = y[b * DY + c] + xy + ey;
  __syncthreads();
  float h = y1b[c];
  for (int k = 0; k < DY; k++) h += T[b * DY + k] * y1W[k * DY + c];
  H[tid] = fmaxf(h, 0.f);
  __syncthreads();
  float o = y2b[c];
  for (int k = 0; k < DY; k++) o += H[b * DY + k] * y2W[k * DY + c];
  outy[tid] = o;
}

// ---------------- host launcher ----------------
extern "C" void kernel_launch(void* const* d_in, const int* in_sizes, int n_in,
                              void* d_out, int out_size, void* d_ws, size_t ws_size,
                              hipStream_t stream) {
  const float* X     = (const float*)d_in[0];
  const float* E     = (const float*)d_in[1];
  const float* y     = (const float*)d_in[2];
  const unsigned char* mask = (const unsigned char*)d_in[3];
  const float* qW    = (const float*)d_in[4];
  const float* qb    = (const float*)d_in[5];
  const float* kW    = (const float*)d_in[6];
  const float* kb    = (const float*)d_in[7];
  const float* vW    = (const float*)d_in[8];
  const float* vb    = (const float*)d_in[9];
  const float* emulW = (const float*)d_in[10];
  const float* emulb = (const float*)d_in[11];
  const float* eaddW = (const float*)d_in[12];
  const float* eaddb = (const float*)d_in[13];
  const float* yxmW  = (const float*)d_in[14];
  const float* yxmb  = (const float*)d_in[15];
  const float* yxaW  = (const float*)d_in[16];
  const float* yxab  = (const float*)d_in[17];
  const float* xoutW = (const float*)d_in[18];
  const float* xoutb = (const float*)d_in[19];
  const float* eoutW = (const float*)d_in[20];
  const float* eoutb = (const float*)d_in[21];
  const float* xyW   = (const float*)d_in[22];
  const float* xyb   = (const float*)d_in[23];
  const float* eyW   = (const float*)d_in[24];
  const float* eyb   = (const float*)d_in[25];
  const float* y1W   = (const float*)d_in[26];
  const float* y1b   = (const float*)d_in[27];
  const float* y2W   = (const float*)d_in[28];
  const float* y2b   = (const float*)d_in[29];

  float* out = (float*)d_out;
  float* newX = out;                                   // 768*256
  float* newE = out + (size_t)ROWS * DX;               // 147456*64
  float* outy = out + (size_t)ROWS * DX + (size_t)EROWS * DE;  // 4*64

  // workspace layout (floats)
  float* ws = (float*)d_ws;
  float* Q      = ws;                          // 196608
  float* Km     = Q   + (size_t)ROWS * DX;     // 196608
  float* Vm     = Km  + (size_t)ROWS * DX;     // 196608
  float* wV     = Vm  + (size_t)ROWS * DX;     // 196608
  float* scores = wV  + (size_t)ROWS * DX;     // 1179648 (softmax in place)
  float* yx1    = scores + (size_t)EROWS * NHEAD;  // 1024
  float* yx2    = yx1 + BS * DX;               // 1024
  float* zx     = yx2 + BS * DX;               // 4096
  float* ze     = zx + BS * 4 * DX;            // 1024

  // 1) Q, K, V
  qkv_kernel<<<dim3(ROWS / 16, DX / 16, 3), 32, 0, stream>>>(
      X, qW, qb, kW, kb, vW, vb, mask, Q, Km, Vm);

  // 2) fused edge pipeline: TDM E-tile -> E1/E2 -> Y (LDS) -> scores + newE
  edge_kernel<<<dim3(EROWS / 16), 32, 0, stream>>>(
      E, Q, Km, emulW, emulb, eaddW, eaddb, eoutW, eoutb, mask, scores, newE);

  // 3) softmax over j
  softmax_kernel<<<dim3(BS * NN * NHEAD), 32, 0, stream>>>(scores);

  // 4) weighted_V
  wv_kernel<<<dim3(NN / 16, DF / 16, BS * NHEAD), 32, 0, stream>>>(scores, Vm, wV);

  // 5) yx1 / yx2
  yx_kernel<<<dim3(8), 256, 0, stream>>>(y, yxaW, yxab, yxmW, yxmb, yx1, yx2);

  // 6) newX
  newx_kernel<<<dim3(ROWS / 16, DX / 16), 32, 0, stream>>>(
      yx1, yx2, wV, xoutW, xoutb, mask, newX);

  // 7) X stats
  zx_kernel<<<dim3(BS * DX), 32, 0, stream>>>(X, zx);

  // 8) E stats
  ze_kernel<<<dim3(BS * DE), 256, 0, stream>>>(E, ze);

  // 9) y head
  yfinal_kernel<<<dim3(1), 256, 0, stream>>>(
      y, zx, ze, xyW, xyb, eyW, eyb, y1W, y1b, y2W, y2b, outy);
}